// CIN_12532714570280
// MI455X (gfx1250) — compile-verified
//
#include <hip/hip_runtime.h>

// ---------------------------------------------------------------------------
// CIN (xDeepFM) on gfx1250 using V_WMMA_F32_16X16X4_F32.
//   BATCH=1024, FIELDS=39, DIM=32, L0_OUT=256, L1_OUT=128
// Compute-bound (~67 GFLOP over ~10MB): Z (outer products) is synthesized
// per-tile from LDS, never touching HBM. fp32 WMMA keeps reference precision.
// ---------------------------------------------------------------------------

typedef __attribute__((ext_vector_type(2))) float v2f;
typedef __attribute__((ext_vector_type(8))) float v8f;

#define BATCH   1024
#define FIELDS  39
#define DIM     32
#define K0      (FIELDS * FIELDS)     // 1521
#define K0P     1524                  // padded to multiple of 4
#define O0      256
#define K1      (FIELDS * 128)        // 4992 (already % 4 == 0)
#define O1      128
#define XSZ     (FIELDS * DIM)        // 1248 floats per batch element

static __device__ __forceinline__ v8f wmma4(v2f a, v2f b, v8f c) {
    // D(16x16,f32) = A(16x4,f32) x B(4x16,f32) + C
    return __builtin_amdgcn_wmma_f32_16x16x4_f32(
        /*neg_a=*/false, a, /*neg_b=*/false, b,
        /*c_mod=*/(short)0, c, /*reuse_a=*/false, /*reuse_b=*/false);
}

static __device__ __forceinline__ float ldsf(const char* base, int byteOff) {
    return *(const float*)(base + byteOff);
}

// Zero-pad w0 (256 x 1521) -> w0p (256 x 1524) so the K-loop is uniform.
__global__ __launch_bounds__(256) void cin_pack_w0(
    const float* __restrict__ w0, float* __restrict__ w0p) {
    int idx = blockIdx.x * 256 + threadIdx.x;
    if (idx >= O0 * K0P) return;
    int o = idx / K0P;
    int k = idx - o * K0P;
    w0p[idx] = (k < K0) ? w0[o * K0 + k] : 0.0f;
}

// ---------------------------------------------------------------------------
// Layer 0:  h0[b,o,d] = relu(sum_k w0[o,k] * x[b,k/39,d]*x[b,k%39,d] + b0[o])
//   rows   0..127 -> h1a (workspace, feeds layer 1)
//   rows 128..255 -> sum over d -> out[b, 0..127]
// One block per batch element. 8 waves; wave w owns O rows [32w, 32w+32).
// (m,n) = (k/39, k%39) maintained incrementally as LDS *byte* offsets
// (no per-iteration shift to bytes): per quad n advances by 4 (<39), so at
// most one wrap -> cmp + 2 selects.
// ---------------------------------------------------------------------------
__global__ __launch_bounds__(256) void cin_layer0(
    const float* __restrict__ x, const float* __restrict__ w0p,
    const float* __restrict__ b0, float* __restrict__ h1a,
    float* __restrict__ out) {
    __shared__ float xs[40 * DIM];   // 39 real rows + 1 zero row (pad-k safety)

    const int b   = blockIdx.x;
    const int tid = threadIdx.x;
    for (int i = tid; i < 40 * DIM; i += 256)
        xs[i] = (i < XSZ) ? x[b * XSZ + i] : 0.0f;
    __syncthreads();

    const int wave = tid >> 5;
    const int lane = tid & 31;
    const int half = lane >> 4;      // 0: K pair {0,1}; 1: K pair {2,3}
    const int l16  = lane & 15;
    const int orow = wave * 32;

    const float* wr0 = w0p + (orow + l16) * K0P + 2 * half;       // A tile 0
    const float* wr1 = wr0 + 16 * K0P;                            // A tile 1
    const int d0 = l16, d1 = l16 + 16;

    // Incremental (m,n) as BYTE offsets into xs, with +d0 folded in.
    const char* xb = (const char*)xs;
    int mA = d0 * 4;                           // (m*DIM + d0)*4     (k = k0+2h)
    int nA = (2 * half * DIM + d0) * 4;        // (n*DIM + d0)*4
    int mB = d0 * 4;                           // (k+1)
    int nB = ((2 * half + 1) * DIM + d0) * 4;
    const int thresh = (FIELDS * DIM + d0) * 4;   // wrap when n >= FIELDS

    v8f acc[2][2] = {};   // [o-tile][d-tile]

    for (int k0 = 0; k0 < K0P; k0 += 4) {
        v2f a0 = *(const v2f*)(wr0 + k0);
        v2f a1 = *(const v2f*)(wr1 + k0);

        v2f bt0, bt1;                                  // B for d 0-15 / 16-31
        bt0.x = ldsf(xb, mA)      * ldsf(xb, nA);
        bt0.y = ldsf(xb, mB)      * ldsf(xb, nB);
        bt1.x = ldsf(xb, mA + 64) * ldsf(xb, nA + 64);
        bt1.y = ldsf(xb, mB + 64) * ldsf(xb, nB + 64);

        // advance k by 4: n += 4 rows (one possible wrap)
        nA += 4 * DIM * 4;
        {   const bool w = (nA >= thresh);
            nA -= w ? FIELDS * DIM * 4 : 0;  mA += w ? DIM * 4 : 0; }
        nB += 4 * DIM * 4;
        {   const bool w = (nB >= thresh);
            nB -= w ? FIELDS * DIM * 4 : 0;  mB += w ? DIM * 4 : 0; }

        acc[0][0] = wmma4(a0, bt0, acc[0][0]);
        acc[0][1] = wmma4(a0, bt1, acc[0][1]);
        acc[1][0] = wmma4(a1, bt0, acc[1][0]);
        acc[1][1] = wmma4(a1, bt1, acc[1][1]);
    }

    if (wave < 4) {
        // rows 0..127: store relu(h0) as h1a[b][row][d]
        float* hp = h1a + b * (O1 * DIM);
#pragma unroll
        for (int ot = 0; ot < 2; ++ot) {
#pragma unroll
            for (int j = 0; j < 8; ++j) {
                const int row  = orow + 16 * ot + j + 8 * half;
                const float bs = b0[row];
                hp[row * DIM + d0] = fmaxf(acc[ot][0][j] + bs, 0.0f);
                hp[row * DIM + d1] = fmaxf(acc[ot][1][j] + bs, 0.0f);
            }
        }
    } else {
        // rows 128..255: relu then sum over d -> out[b, row-128]
#pragma unroll
        for (int ot = 0; ot < 2; ++ot) {
#pragma unroll
            for (int j = 0; j < 8; ++j) {
                const int row  = orow + 16 * ot + j + 8 * half;
                const float bs = b0[row];
                float v = fmaxf(acc[ot][0][j] + bs, 0.0f) +
                          fmaxf(acc[ot][1][j] + bs, 0.0f);
                v += __shfl_xor(v, 1);
                v += __shfl_xor(v, 2);
                v += __shfl_xor(v, 4);
                v += __shfl_xor(v, 8);       // reduce within each 16-lane half
                if (l16 == 0) out[b * 256 + (row - 128)] = v;
            }
        }
    }
}

// ---------------------------------------------------------------------------
// Layer 1:  h1[b,o,d] = relu(sum_k w1[o,k] * x[b,k>>7,d]*h1a[b,k&127,d] + b1[o])
//   out[b, 128+o] = sum_d h1[b,o,d]
// One block per batch element. 8 waves; wave w owns O rows [16w, 16w+16).
// Same incremental byte-offset scheme (wrap at n==128).
// ---------------------------------------------------------------------------
__global__ __launch_bounds__(256) void cin_layer1(
    const float* __restrict__ x, const float* __restrict__ h1a,
    const float* __restrict__ w1, const float* __restrict__ b1,
    float* __restrict__ out) {
    __shared__ float xs[XSZ];            // 4992 B
    __shared__ float hs[O1 * DIM];       // 16 KB

    const int b   = blockIdx.x;
    const int tid = threadIdx.x;
    for (int i = tid; i < XSZ; i += 256)      xs[i] = x[b * XSZ + i];
    for (int i = tid; i < O1 * DIM; i += 256) hs[i] = h1a[b * O1 * DIM + i];
    __syncthreads();

    const int wave = tid >> 5;
    const int lane = tid & 31;
    const int half = lane >> 4;
    const int l16  = lane & 15;
    const int orow = wave * 16;

    const float* wr = w1 + (orow + l16) * K1 + 2 * half;
    const int d0 = l16;

    const char* xbb = (const char*)xs;
    const char* hbb = (const char*)hs;
    int mA = d0 * 4;                           // ((k>>7)*DIM + d0)*4
    int nA = (2 * half * DIM + d0) * 4;        // ((k&127)*DIM + d0)*4
    int mB = d0 * 4;                           // (k+1)
    int nB = ((2 * half + 1) * DIM + d0) * 4;
    const int thresh = (O1 * DIM + d0) * 4;    // wrap when n >= 128

    v8f acc0 = {}, acc1 = {};

    for (int k0 = 0; k0 < K1; k0 += 4) {
        v2f a = *(const v2f*)(wr + k0);

        v2f bt0, bt1;
        bt0.x = ldsf(xbb, mA)      * ldsf(hbb, nA);
        bt0.y = ldsf(xbb, mB)      * ldsf(hbb, nB);
        bt1.x = ldsf(xbb, mA + 64) * ldsf(hbb, nA + 64);
        bt1.y = ldsf(xbb, mB + 64) * ldsf(hbb, nB + 64);

        nA += 4 * DIM * 4;
        {   const bool w = (nA >= thresh);
            nA -= w ? O1 * DIM * 4 : 0;  mA += w ? DIM * 4 : 0; }
        nB += 4 * DIM * 4;
        {   const bool w = (nB >= thresh);
            nB -= w ? O1 * DIM * 4 : 0;  mB += w ? DIM * 4 : 0; }

        acc0 = wmma4(a, bt0, acc0);
        acc1 = wmma4(a, bt1, acc1);
    }

#pragma unroll
    for (int j = 0; j < 8; ++j) {
        const int row  = orow + j + 8 * half;
        const float bs = b1[row];
        float v = fmaxf(acc0[j] + bs, 0.0f) + fmaxf(acc1[j] + bs, 0.0f);
        v += __shfl_xor(v, 1);
        v += __shfl_xor(v, 2);
        v += __shfl_xor(v, 4);
        v += __shfl_xor(v, 8);
        if (l16 == 0) out[b * 256 + 128 + row] = v;
    }
}

// ---------------------------------------------------------------------------
extern "C" void kernel_launch(void* const* d_in, const int* in_sizes, int n_in,
                              void* d_out, int out_size, void* d_ws, size_t ws_size,
                              hipStream_t stream) {
    const float* x  = (const float*)d_in[0];   // (1024, 39, 32)
    const float* w0 = (const float*)d_in[1];   // (256, 1521)
    const float* b0 = (const float*)d_in[2];   // (256,)
    const float* w1 = (const float*)d_in[3];   // (128, 4992)
    const float* b1 = (const float*)d_in[4];   // (128,)
    float* out = (float*)d_out;                // (1024, 256)

    float* w0p  = (float*)d_ws;                       // 256*1524 floats
    float* h1a  = w0p + (size_t)O0 * K0P;             // 1024*128*32 floats

    cin_pack_w0<<<(O0 * K0P + 255) / 256, 256, 0, stream>>>(w0, w0p);
    cin_layer0 <<<BATCH, 256, 0, stream>>>(x, w0p, b0, h1a, out);
    cin_layer1 <<<BATCH, 256, 0, stream>>>(x, h1a, w1, b1, out);
}